// dynamic_graph_module_28303834480931
// MI455X (gfx1250) — compile-verified
//
#include <hip/hip_runtime.h>

typedef float v2f __attribute__((ext_vector_type(2)));
typedef float v8f __attribute__((ext_vector_type(8)));

// Problem constants (from reference)
#define SEQ   2048
#define BH    16      // bsz * nhead
#define HD    64      // head_dim
#define PD    16      // proj_dim

// -------------------------------------------------------------------------
// Kernel 1a: build xhT (BH, HD, SEQ) fp32 from x (SEQ, BSZ, D_MODEL).
// Index algebra for the reshape chain:
//   xh[bh][t][k] = x[s_x][b_x][d_x],
//   b_x = t>=1024, s_x = 2*(t%1024) + (bh>>3), d_x = (bh&7)*64 + k
// We store it transposed (d-major) so the attention kernel can do b64 loads
// of K-consecutive elements for the WMMA B operand.
// -------------------------------------------------------------------------
__global__ __launch_bounds__(256) void build_xhT(const float* __restrict__ x,
                                                 float* __restrict__ xhT) {
  int idx = blockIdx.x * blockDim.x + threadIdx.x;   // (bh*64 + k)*2048 + t
  int t  = idx & 2047;
  int bk = idx >> 11;
  int k  = bk & 63;
  int bh = bk >> 6;
  int b_x = t >> 10;
  int s_x = 2 * (t & 1023) + (bh >> 3);
  int d_x = (bh & 7) * 64 + k;
  xhT[idx] = x[s_x * 1024 + b_x * 512 + d_x];
}

// -------------------------------------------------------------------------
// Kernel 1b: feat1/feat2 (BH, SEQ, PD) = relu(xh @ w^T + b).
// One thread per (bh, q, t), q<16 -> feat1[p=q], q>=16 -> feat2[p=q-16].
// Consecutive threads share (bh,q) and walk t -> coalesced xhT reads.
// -------------------------------------------------------------------------
__global__ __launch_bounds__(256) void build_feats(
    const float* __restrict__ xhT,
    const float* __restrict__ w1, const float* __restrict__ b1,
    const float* __restrict__ w2, const float* __restrict__ b2,
    float* __restrict__ feat1, float* __restrict__ feat2) {
  int idx = blockIdx.x * blockDim.x + threadIdx.x;   // (bh*32 + q)*2048 + t
  int t  = idx & 2047;
  int q  = (idx >> 11) & 31;
  int bh = idx >> 16;
  const float* w; float bias; float* dst; int p;
  if (q < 16) { p = q;      w = w1 + p * HD; bias = b1[p]; dst = feat1; }
  else        { p = q - 16; w = w2 + p * HD; bias = b2[p]; dst = feat2; }
  const float* xr = xhT + (long)bh * HD * SEQ + t;
  float s = bias;
#pragma unroll 8
  for (int k = 0; k < HD; ++k) s += xr[(long)k * SEQ] * w[k];
  dst[((long)bh * SEQ + t) * PD + p] = fmaxf(s, 0.0f);
}

// -------------------------------------------------------------------------
// Kernel 2: fused  A = feat1@feat2^T -> threshold -> store A ->
//           unnormalized masked softmax -> out = alpha @ xh.
// One wave32 per (b, 16-row s-tile). Flash-style accumulation with the
// max-subtraction dropped (softmax is shift invariant; |A| is small).
// Uses V_WMMA_F32_16X16X4_F32 for both GEMMs.
//   C/D layout: c[v], lane l -> row M=v+8*(l>>4), col N=l&15
//   A layout  : lane l -> row M=l&15, a.x=K(2h), a.y=K(2h+1), h=l>>4
//   B layout  : lane l -> col N=l&15, b.x=K(2h), b.y=K(2h+1)
// -------------------------------------------------------------------------
__global__ __launch_bounds__(32) void attn_fused(
    const float* __restrict__ xhT,    // (BH, HD, SEQ)
    const float* __restrict__ feat1,  // (BH, SEQ, PD)
    const float* __restrict__ feat2,  // (BH, SEQ, PD)
    float* __restrict__ outO,         // (BH, SEQ, HD)
    float* __restrict__ outA) {       // (BH, SEQ, SEQ)
  __shared__ float p_lds[16 * 16];    // masked-exp tile, single wave owns it

  const int lane = threadIdx.x;
  const int n = lane & 15;            // N / column index
  const int h = lane >> 4;            // half-wave select
  const int b  = blockIdx.x >> 7;     // 16 heads*batch
  const int s0 = (blockIdx.x & 127) << 4;

  // A operand of matmul 1: feat1 rows s0..s0+15, all 16 p. 4 K-slices, each a
  // b64 load of two consecutive p values for this lane's row m = n.
  v2f a1[4];
  const float* f1row = feat1 + ((long)(b * SEQ + s0 + n) * PD);
#pragma unroll
  for (int k = 0; k < 4; ++k)
    a1[k] = *(const v2f*)(f1row + 4 * k + 2 * h);

  v8f acc[4];                          // out tile accumulators (4 d-tiles)
#pragma unroll
  for (int dt = 0; dt < 4; ++dt)
#pragma unroll
    for (int v = 0; v < 8; ++v) acc[dt][v] = 0.0f;

  float lsum[8];                       // per-lane partial softmax denominators
#pragma unroll
  for (int v = 0; v < 8; ++v) lsum[v] = 0.0f;

  const float* f2base = feat2 + (long)b * SEQ * PD;
  const float* xbase  = xhT   + (long)b * HD * SEQ;
  float* Abase = outA + (long)b * SEQ * SEQ;

  for (int t0 = 0; t0 < SEQ; t0 += 16) {
    // ---- matmul 1: 16x16 score tile, K=16 in four x4 steps -------------
    v8f c;
#pragma unroll
    for (int v = 0; v < 8; ++v) c[v] = 0.0f;
    const float* f2row = f2base + (long)(t0 + n) * PD + 2 * h;
#pragma unroll
    for (int k = 0; k < 4; ++k) {
      v2f b2 = *(const v2f*)(f2row + 4 * k);
      c = __builtin_amdgcn_wmma_f32_16x16x4_f32(false, a1[k], false, b2,
                                                (short)0, c, false, false);
    }
    if (t0 + 16 < SEQ)  // global_prefetch_b8 of next feat2 chunk
      __builtin_prefetch(f2base + (long)(t0 + 16 + n) * PD, 0, 1);

    // ---- threshold, emit A, exp, masked weights ------------------------
#pragma unroll
    for (int v = 0; v < 8; ++v) {
      float a   = c[v];
      float ath = (a < 0.2f) ? 0.0f : a;
      Abase[(long)(s0 + v + 8 * h) * SEQ + t0 + n] = ath;
      float e = __expf(ath);           // exp(0)=1 for masked entries
      lsum[v] += e;                    // denominator counts everything
      float pm = (ath == 0.0f) ? 0.0f : e;   // numerator is masked
      p_lds[(v + 8 * h) * 16 + n] = pm;
    }
    __builtin_amdgcn_wave_barrier();   // LDS is in-order intra-wave; pin sched

    // ---- matmul 2: acc += p_tile(16x16) @ xh_chunk(16x64) --------------
#pragma unroll
    for (int k = 0; k < 4; ++k) {
      v2f ap = *(const v2f*)(&p_lds[n * 16 + 4 * k + 2 * h]);
#pragma unroll
      for (int dt = 0; dt < 4; ++dt) {
        v2f bx = *(const v2f*)(xbase + (long)(dt * 16 + n) * SEQ +
                               t0 + 4 * k + 2 * h);
        acc[dt] = __builtin_amdgcn_wmma_f32_16x16x4_f32(
            false, ap, false, bx, (short)0, acc[dt], false, false);
      }
    }
    __builtin_amdgcn_wave_barrier();
  }

  // ---- epilogue: reduce denominators across the 16-lane halves, divide ----
#pragma unroll
  for (int v = 0; v < 8; ++v) {
    float s = lsum[v];
    s += __shfl_xor(s, 1);
    s += __shfl_xor(s, 2);
    s += __shfl_xor(s, 4);
    s += __shfl_xor(s, 8);             // stays inside each 16-lane half
    float inv = 1.0f / s;
#pragma unroll
    for (int dt = 0; dt < 4; ++dt)
      outO[(long)(b * SEQ + s0 + v + 8 * h) * HD + dt * 16 + n] =
          acc[dt][v] * inv;
  }
}

// -------------------------------------------------------------------------
extern "C" void kernel_launch(void* const* d_in, const int* in_sizes, int n_in,
                              void* d_out, int out_size, void* d_ws,
                              size_t ws_size, hipStream_t stream) {
  const float* x     = (const float*)d_in[0];
  const float* fc1_w = (const float*)d_in[1];
  const float* fc1_b = (const float*)d_in[2];
  const float* fc2_w = (const float*)d_in[3];
  const float* fc2_b = (const float*)d_in[4];

  // workspace layout (fp32): xhT 8 MB | feat1 2 MB | feat2 2 MB
  float* xhT   = (float*)d_ws;
  float* feat1 = xhT + (size_t)BH * HD * SEQ;          // +2,097,152
  float* feat2 = feat1 + (size_t)BH * SEQ * PD;        // +524,288

  float* outO = (float*)d_out;                         // (16,2048,64)
  float* outA = outO + (size_t)BH * SEQ * HD;          // (16,2048,2048)

  build_xhT<<<(BH * HD * SEQ) / 256, 256, 0, stream>>>(x, xhT);
  build_feats<<<(BH * 2 * PD * SEQ) / 256, 256, 0, stream>>>(
      xhT, fc1_w, fc1_b, fc2_w, fc2_b, feat1, feat2);
  attn_fused<<<BH * (SEQ / 16), 32, 0, stream>>>(xhT, feat1, feat2, outO, outA);
}